// simpleDFN_231928234098
// MI455X (gfx1250) — compile-verified
//
#include <hip/hip_runtime.h>

typedef __attribute__((ext_vector_type(2))) float v2f;
typedef __attribute__((ext_vector_type(8))) float v8f;

#define HW     120
#define CH     256
#define OCH    256
#define NB     8
#define PLANE  (HW * HW)          // 14400
#define CCHUNK 64
#define MIDR   18                 // 16 output rows + 2 halo
#define MIDC   18                 // 16 output cols + 2 halo
#define MIDRC  (MIDR * MIDC)      // 324

// ---------------------------------------------------------------------------
// Kernel 1: adaptive avg-pool 120x120 -> 3x3 per (b,c) plane.
// One 256-thread block per plane; 40x40 cells, strided accumulate + LDS tree.
// ---------------------------------------------------------------------------
__global__ __launch_bounds__(256) void dfn_pool(const float* __restrict__ y,
                                                float* __restrict__ yp) {
  const int bc = blockIdx.x;                       // b*C + c
  const float* p = y + (size_t)bc * PLANE;
  float acc[9] = {0.f,0.f,0.f,0.f,0.f,0.f,0.f,0.f,0.f};
  for (int idx = threadIdx.x; idx < PLANE; idx += 256) {
    int h = idx / HW, w = idx - h * HW;
    acc[(h / 40) * 3 + (w / 40)] += p[idx];
  }
  __shared__ float red[256];
  for (int cell = 0; cell < 9; ++cell) {
    red[threadIdx.x] = acc[cell];
    __syncthreads();
    for (int s = 128; s > 0; s >>= 1) {
      if (threadIdx.x < s) red[threadIdx.x] += red[threadIdx.x + s];
      __syncthreads();
    }
    if (threadIdx.x == 0) yp[bc * 9 + cell] = red[0] * (1.0f / 1600.0f);
    __syncthreads();
  }
}

// ---------------------------------------------------------------------------
// Kernel 2: per-sample kernel generation.
// dk[b][o][ij] = sum_c gen_w[o][c] * yp[b][c][ij] + gen_b[o]   (18432 dots)
// ---------------------------------------------------------------------------
__global__ __launch_bounds__(256) void dfn_gen(const float* __restrict__ yp,
                                               const float* __restrict__ gw,
                                               const float* __restrict__ gb,
                                               float* __restrict__ dk) {
  const int idx = blockIdx.x * 256 + threadIdx.x;  // b*2304 + o*9 + ij
  if (idx >= NB * CH * 9) return;
  const int ij = idx % 9;
  const int o  = (idx / 9) % CH;
  const int b  = idx / (9 * CH);
  const float* wrow = gw + (size_t)o * CH;
  const float* ycol = yp + (size_t)b * CH * 9 + ij;
  float s = gb[o];
#pragma unroll 4
  for (int c = 0; c < CH; ++c) s += wrow[c] * ycol[c * 9];
  dk[idx] = s;
}

// ---------------------------------------------------------------------------
// Kernel 3: fused dynamic-depthwise-3x3 + dense 3x3 conv (implicit GEMM).
// Block = (batch, 16x16 spatial tile, 128-wide OC half). 8 waves.
// Per 64-channel chunk: depthwise into LDS mid[64][18][18], then WMMA K-loop
// over K = 9 taps * 64 channels with V_WMMA_F32_16X16X4_F32.
// Wave w owns M-frag row w (16 OC) x 16 N-frags (N = hy*16 + wx).
// ---------------------------------------------------------------------------
__global__ __launch_bounds__(256) void dfn_fuse(const float* __restrict__ x,
                                                const float* __restrict__ dk,
                                                const float* __restrict__ fw,
                                                const float* __restrict__ fb,
                                                float* __restrict__ out) {
  __shared__ float mid[CCHUNK * MIDRC];            // 82944 B, [c][r][col]

  const int tid  = threadIdx.x;
  const int lane = tid & 31;
  const int wid  = tid >> 5;                       // 0..7
  const int lm   = lane & 15;                      // M-row / N-col within frag
  const int lo   = (lane >> 4) << 1;               // K sub-offset: 0 | 2

  const int wt    = blockIdx.x & 7;
  const int mhalf = blockIdx.x >> 3;
  const int h0 = blockIdx.y * 16;
  const int w0 = wt * 16;
  const int b  = blockIdx.z;
  const int mbase = mhalf * 128 + wid * 16;

  v8f acc[16];
  const v8f vzero = {0.f,0.f,0.f,0.f,0.f,0.f,0.f,0.f};
#pragma unroll
  for (int nf = 0; nf < 16; ++nf) acc[nf] = vzero;

  // A-matrix row for this lane (fuse_w is [OC][C][3][3], row stride C*9)
  const float* fwl = fw + (size_t)(mbase + lm) * (CH * 9);

  for (int cc0 = 0; cc0 < CH; cc0 += CCHUNK) {
    // ---- depthwise: build mid tile (incl. halo) for this channel chunk ----
    for (int idx = tid; idx < CCHUNK * MIDRC; idx += 256) {
      const int cl  = idx / MIDRC;
      const int rem = idx - cl * MIDRC;
      const int r   = rem / MIDC;
      const int col = rem - r * MIDC;
      const int h = h0 - 1 + r;
      const int w = w0 - 1 + col;
      float v = 0.0f;
      if (h >= 0 && h < HW && w >= 0 && w < HW) {
        const size_t chan = (size_t)(b * CH + cc0 + cl);
        const float* kp = dk + chan * 9;
        const float* xp = x + chan * PLANE;
#pragma unroll
        for (int i = 0; i < 3; ++i) {
          const int hh = h - 1 + i;
          if (hh < 0 || hh >= HW) continue;
#pragma unroll
          for (int j = 0; j < 3; ++j) {
            const int ww = w - 1 + j;
            if (ww < 0 || ww >= HW) continue;
            v += kp[i * 3 + j] * xp[hh * HW + ww];
          }
        }
      }
      mid[idx] = v;
    }
    __syncthreads();

    // ---- implicit GEMM: K = tap-major (9 taps x 64 channels) ----
    for (int tap = 0; tap < 9; ++tap) {
      const int ti = tap / 3, tj = tap % 3;
      const float* fwt  = fwl + (size_t)cc0 * 9 + tap;   // + c*9 below
      const int    moff = ti * MIDC + tj + lm;           // lane B-col offset
      for (int cb = 0; cb < CCHUNK; cb += 4) {
        const int c0 = cb + lo;                          // this half's K base
        v2f a;                                           // A frag: K=c0, c0+1
        a.x = fwt[c0 * 9];
        a.y = fwt[c0 * 9 + 9];
        const float* m0p = mid + c0 * MIDRC + moff;
        const float* m1p = m0p + MIDRC;
#pragma unroll
        for (int nf = 0; nf < 16; ++nf) {
          v2f bb;                                        // B frag: N=n0+lm
          bb.x = m0p[nf * MIDC];
          bb.y = m1p[nf * MIDC];
          acc[nf] = __builtin_amdgcn_wmma_f32_16x16x4_f32(
              false, a, false, bb, (short)0, acc[nf], false, false);
        }
      }
    }
    __syncthreads();
  }

  // ---- epilogue: bias + store (C frag: VGPR v -> M=v (+8 upper half)) ----
  const int mo = (lane < 16) ? 0 : 8;
  float bias[8];
#pragma unroll
  for (int v = 0; v < 8; ++v) bias[v] = fb[mbase + v + mo];

  const int w = w0 + lm;
  if (w < HW) {
#pragma unroll
    for (int nf = 0; nf < 16; ++nf) {
      const int h = h0 + nf;
      if (h >= HW) break;
#pragma unroll
      for (int v = 0; v < 8; ++v) {
        const int oc = mbase + v + mo;
        out[(((size_t)b * OCH + oc) * HW + h) * HW + w] = acc[nf][v] + bias[v];
      }
    }
  }
}

// ---------------------------------------------------------------------------
extern "C" void kernel_launch(void* const* d_in, const int* in_sizes, int n_in,
                              void* d_out, int out_size, void* d_ws, size_t ws_size,
                              hipStream_t stream) {
  const float* x  = (const float*)d_in[0];
  const float* y  = (const float*)d_in[1];
  const float* gw = (const float*)d_in[2];
  const float* gb = (const float*)d_in[3];
  const float* fw = (const float*)d_in[4];
  const float* fb = (const float*)d_in[5];
  float* out = (float*)d_out;

  float* yp = (float*)d_ws;            // 8*256*9 = 18432 floats
  float* dk = yp + NB * CH * 9;        // 8*256*9 = 18432 floats

  dfn_pool<<<NB * CH, 256, 0, stream>>>(y, yp);
  dfn_gen<<<(NB * CH * 9 + 255) / 256, 256, 0, stream>>>(yp, gw, gb, dk);

  dim3 grid(16 /* 8 w-tiles * 2 OC halves */, 8 /* h-tiles */, NB);
  dfn_fuse<<<grid, 256, 0, stream>>>(x, dk, fw, fb, out);
}